// ParallelHashEmbedder_5007931867220
// MI455X (gfx1250) — compile-verified
//
#include <hip/hip_runtime.h>

// ---------------------------------------------------------------------------
// Instant-NGP multires hash encoding for MI455X (gfx1250).
// Memory/gather-bound: ~134 MB HBM floor (~6us @ 23.3 TB/s); 67M 8B L2 gathers.
// WMMA does not apply (per-point 8-weight dots, no shared operand matrix).
// CDNA5-specific path used: Tensor Data Mover stages each block's coordinate
// slab into LDS (TENSORcnt / s_wait_tensorcnt), freeing the VMEM pipe for the
// 128 table gathers per point. Output transposed through LDS for coalescing.
// ---------------------------------------------------------------------------

#define N_LEVELS      16
#define N_FEAT        2
#define LOG2_HASHMAP  19
#define HASHMAP_SIZE  (1u << LOG2_HASHMAP)
#define BLOCK         256
#define OUT_STRIDE    33   // 32 outputs + 1 pad float -> conflict-free LDS transpose

typedef __attribute__((ext_vector_type(4))) unsigned int u32x4;
typedef __attribute__((ext_vector_type(8))) int          i32x8;
typedef __attribute__((ext_vector_type(4))) int          i32x4;

struct SMem {
  // MUST be first member: the TDM descriptor targets LDS byte offset 0.
  float coords[BLOCK * 3];
  float outT[BLOCK * OUT_STRIDE];
};

__global__ __launch_bounds__(BLOCK)
void ngp_hash_embed_kernel(const float* __restrict__ x,
                           const float* __restrict__ table,
                           const float* __restrict__ box_min,
                           const float* __restrict__ box_max,
                           float* __restrict__ out,
                           int B) {
  __shared__ SMem sm;

  const int tid       = (int)threadIdx.x;
  const int blockBase = (int)blockIdx.x * BLOCK;
  const int remi      = B - blockBase;
  const int rem       = remi < BLOCK ? remi : BLOCK;      // points in this block
  const unsigned nElem = (unsigned)rem * 3u;              // floats to stage

  // ---- Stage coordinates into LDS via the Tensor Data Mover (wave 0) ----
#if __has_builtin(__builtin_amdgcn_tensor_load_to_lds)
  if (tid < 32) {
    unsigned long long ga =
        (unsigned long long)(const void*)(x + (size_t)blockBase * 3);
    // D# group 0 (128b): count=1, lds_addr=0 (sm.coords), global_addr, type=2
    u32x4 g0;
    g0.x = 1u;                                            // count = 1 (valid)
    g0.y = 0u;                                            // lds_addr (bytes)
    g0.z = (unsigned)(ga & 0xFFFFFFFFull);                // global_addr[31:0]
    g0.w = ((unsigned)(ga >> 32) & 0x01FFFFFFu)           // global_addr[56:32]
           | (2u << 30);                                  // type = 2 ("image")
    // D# group 1 (256b): data_size=4B, 2D tile = nElem x 1
    i32x8 g1;
    g1[0] = (int)(2u << 16);                              // data_size=2 (4B)
    g1[1] = (int)((nElem & 0xFFFFu) << 16);               // tensor_dim0[15:0]
    g1[2] = (int)((nElem >> 16) | (1u << 16));            // dim0[31:16] | tensor_dim1=1
    g1[3] = (int)((nElem & 0xFFFFu) << 16);               // tile_dim0 = nElem
    g1[4] = 1;                                            // tile_dim1 = 1
    g1[5] = (int)nElem;                                   // tensor_dim0_stride
    g1[6] = 0;
    g1[7] = 0;
    i32x4 g2 = {0, 0, 0, 0};                              // 3D+ dims unused
    i32x4 g3 = {0, 0, 0, 0};
    i32x8 g4 = {0, 0, 0, 0, 0, 0, 0, 0};                  // unused 5th slot (NULL)
    __builtin_amdgcn_tensor_load_to_lds(g0, g1, g2, g3, g4, /*cpol=*/0);
#if __has_builtin(__builtin_amdgcn_s_wait_tensorcnt)
    __builtin_amdgcn_s_wait_tensorcnt(0);
#else
    asm volatile("s_wait_tensorcnt 0x0" ::: "memory");
#endif
  }
#else
  // Fallback: cooperative coalesced staging.
  for (unsigned i = (unsigned)tid; i < nElem; i += BLOCK)
    sm.coords[i] = x[(size_t)blockBase * 3 + i];
#endif
  __syncthreads();

  const bool active = tid < rem;

  // Reciprocal box widths hoisted: 3 divides per thread total.
  const float bm0 = box_min[0], bm1 = box_min[1], bm2 = box_min[2];
  const float iw0 = 1.0f / (box_max[0] - bm0);
  const float iw1 = 1.0f / (box_max[1] - bm1);
  const float iw2 = 1.0f / (box_max[2] - bm2);

  if (active) {
    const float px = sm.coords[tid * 3 + 0];
    const float py = sm.coords[tid * 3 + 1];
    const float pz = sm.coords[tid * 3 + 2];
    const float2* __restrict__ tab = (const float2*)table;

    // floor(16 * (512/16)^(l/15)) for l = 0..15 (standard NGP levels)
    constexpr float kRes[N_LEVELS] = {16.f, 20.f, 25.f, 32.f, 40.f, 50.f,
                                      64.f, 80.f, 101.f, 128.f, 161.f, 203.f,
                                      256.f, 322.f, 406.f, 512.f};
    const unsigned m = HASHMAP_SIZE - 1u;

#pragma unroll
    for (int l = 0; l < N_LEVELS; ++l) {
      const float r  = kRes[l];
      const float sx = r * iw0, sy = r * iw1, sz = r * iw2;
      const float fx = (px - bm0) * sx;
      const float fy = (py - bm1) * sy;
      const float fz = (pz - bm2) * sz;
      const int ix = (int)fx, iy = (int)fy, iz = (int)fz;
      const float wx = fx - (float)ix;
      const float wy = fy - (float)iy;
      const float wz = fz - (float)iz;

      // NGP spatial hash: primes {1, 2654435761, 805459861}
      const unsigned hx0 = (unsigned)ix;
      const unsigned hy0 = (unsigned)iy * 2654435761u;
      const unsigned hz0 = (unsigned)iz * 805459861u;
      const unsigned hx1 = hx0 + 1u;
      const unsigned hy1 = hy0 + 2654435761u;
      const unsigned hz1 = hz0 + 805459861u;
      const unsigned base = (unsigned)l << LOG2_HASHMAP;

      // 8 independent float2 gathers (global_load_b64, L2-resident table).
      const float2 v000 = tab[(((hx0 ^ hy0) ^ hz0) & m) + base];
      const float2 v001 = tab[(((hx0 ^ hy0) ^ hz1) & m) + base];
      const float2 v010 = tab[(((hx0 ^ hy1) ^ hz0) & m) + base];
      const float2 v011 = tab[(((hx0 ^ hy1) ^ hz1) & m) + base];
      const float2 v100 = tab[(((hx1 ^ hy0) ^ hz0) & m) + base];
      const float2 v101 = tab[(((hx1 ^ hy0) ^ hz1) & m) + base];
      const float2 v110 = tab[(((hx1 ^ hy1) ^ hz0) & m) + base];
      const float2 v111 = tab[(((hx1 ^ hy1) ^ hz1) & m) + base];

      const float omx = 1.0f - wx, omy = 1.0f - wy, omz = 1.0f - wz;

      // feature 0 (reference corner order: i=x, j=y, k=z)
      const float a00 = v000.x * omx + v100.x * wx;
      const float a01 = v001.x * omx + v101.x * wx;
      const float a10 = v010.x * omx + v110.x * wx;
      const float a11 = v011.x * omx + v111.x * wx;
      const float a0  = a00 * omy + a10 * wy;
      const float a1  = a01 * omy + a11 * wy;
      sm.outT[tid * OUT_STRIDE + l] = a0 * omz + a1 * wz;

      // feature 1
      const float b00 = v000.y * omx + v100.y * wx;
      const float b01 = v001.y * omx + v101.y * wx;
      const float b10 = v010.y * omx + v110.y * wx;
      const float b11 = v011.y * omx + v111.y * wx;
      const float b0  = b00 * omy + b10 * wy;
      const float b1  = b01 * omy + b11 * wy;
      sm.outT[tid * OUT_STRIDE + N_LEVELS + l] = b0 * omz + b1 * wz;
    }
  }
  __syncthreads();

  // ---- Coalesced output: out[b*32 + feat*16 + level] via LDS transpose ----
  float* __restrict__ ob = out + (size_t)blockBase * (N_FEAT * N_LEVELS);
  const int total = rem * (N_FEAT * N_LEVELS);
#pragma unroll
  for (int i = 0; i < N_FEAT * N_LEVELS; ++i) {
    const int e = i * BLOCK + tid;      // per wave: fixed row, lane = column
    if (e < total)
      ob[e] = sm.outT[(e >> 5) * OUT_STRIDE + (e & 31)];
  }
}

extern "C" void kernel_launch(void* const* d_in, const int* in_sizes, int n_in,
                              void* d_out, int out_size, void* d_ws,
                              size_t ws_size, hipStream_t stream) {
  const float* x       = (const float*)d_in[0];
  const float* table   = (const float*)d_in[1];
  const float* box_min = (const float*)d_in[2];
  const float* box_max = (const float*)d_in[3];
  float* out = (float*)d_out;

  const int B = in_sizes[0] / 3;
  const int blocks = (B + BLOCK - 1) / BLOCK;
  hipLaunchKernelGGL(ngp_hash_embed_kernel, dim3(blocks), dim3(BLOCK), 0,
                     stream, x, table, box_min, box_max, out, B);
}